// Test_Model_26388279066904
// MI455X (gfx1250) — compile-verified
//
#include <hip/hip_runtime.h>

typedef __attribute__((ext_vector_type(2))) float v2f;
typedef __attribute__((ext_vector_type(8))) float v8f;

// Task meta entry, built on device by the init kernel into d_ws (+64KB).
struct TaskMeta {
  long long outBase;  // output element offset for b=0
  long long bStride;  // output element stride per batch
  int l1, l2;         // input irrep degrees
  int pad0, pad1;     // pad to 32B
};

// ---------------------------------------------------------------------------
// Task decode (init kernel only): 256 tasks enumerate (l1,l2,l3,m3) in
// reference loop order (l1 outer 0..3, l2 0..3, l3=|l1-l2|..l1+l2, m3=0..2l3).
// Output layout matches tuple-of-parts concatenated flat:
//   partBase[l3] + b*(2l3+1)*nfrag*4096 + m3*nfrag*4096 + fragIdx*4096 + p*64+q
// ---------------------------------------------------------------------------
__device__ inline void decodeTask(int r, int& oL1, int& oL2, int& oL3, int& oM3,
                                  long long& outBase, long long& bStride) {
  int nfrag[7] = {0, 0, 0, 0, 0, 0, 0};
  for (int a = 0; a < 4; ++a)
    for (int b = 0; b < 4; ++b) {
      int lo = a > b ? a - b : b - a;
      for (int c = lo; c <= a + b; ++c) nfrag[c]++;
    }
  long long partBase[7];
  long long acc = 0;
  for (int c = 0; c < 7; ++c) {
    partBase[c] = acc;
    acc += 64LL * (2 * c + 1) * nfrag[c] * 4096LL;
  }
  int fragIdx[7] = {0, 0, 0, 0, 0, 0, 0};
  int t = 0;
  oL1 = 0; oL2 = 0; oL3 = 0; oM3 = 0; outBase = 0; bStride = 0;
  for (int a = 0; a < 4; ++a)
    for (int b = 0; b < 4; ++b) {
      int lo = a > b ? a - b : b - a;
      for (int c = lo; c <= a + b; ++c) {
        int fi = fragIdx[c]++;
        int n = 2 * c + 1;
        if (r >= t && r < t + n) {
          int m = r - t;
          oL1 = a; oL2 = b; oL3 = c; oM3 = m;
          outBase = partBase[c] + (long long)m * nfrag[c] * 4096LL +
                    (long long)fi * 4096LL;
          bStride = (long long)n * nfrag[c] * 4096LL;
        }
        t += n;
      }
    }
}

// ---------------------------------------------------------------------------
// Init kernel: Clebsch-Gordan coefficients (Racah, f64) + task meta -> d_ws.
// tab[r*64 + (m1*8+m2)] = C[l1,l2,l3][m1][m2][m3(r)], zero-padded to 8x8.
// ---------------------------------------------------------------------------
__device__ inline double dfact(int n) {
  double f = 1.0;
  for (int i = 2; i <= n; ++i) f *= (double)i;
  return f;
}

__device__ double cg_coef(int j1, int m1, int j2, int m2, int j3, int m3) {
  if (m1 + m2 != m3) return 0.0;
  int dj = j1 - j2; if (dj < 0) dj = -dj;
  if (j3 < dj || j3 > j1 + j2) return 0.0;
  double pref = sqrt((2.0 * j3 + 1.0) * dfact(j3 + j1 - j2) * dfact(j3 - j1 + j2) *
                     dfact(j1 + j2 - j3) / dfact(j1 + j2 + j3 + 1));
  pref *= sqrt(dfact(j3 + m3) * dfact(j3 - m3) * dfact(j1 - m1) * dfact(j1 + m1) *
               dfact(j2 - m2) * dfact(j2 + m2));
  int kmin = 0;
  if (-(j3 - j2 + m1) > kmin) kmin = -(j3 - j2 + m1);
  if (-(j3 - j1 - m2) > kmin) kmin = -(j3 - j1 - m2);
  int kmax = j1 + j2 - j3;
  if (j1 - m1 < kmax) kmax = j1 - m1;
  if (j2 + m2 < kmax) kmax = j2 + m2;
  double s = 0.0;
  for (int k = kmin; k <= kmax; ++k) {
    double d = dfact(k) * dfact(j1 + j2 - j3 - k) * dfact(j1 - m1 - k) *
               dfact(j2 + m2 - k) * dfact(j3 - j2 + m1 + k) * dfact(j3 - j1 - m2 + k);
    s += ((k & 1) ? -1.0 : 1.0) / d;
  }
  return pref * s;
}

__global__ void cg_init_kernel(float* __restrict__ tab, TaskMeta* __restrict__ meta) {
  int r = blockIdx.x;   // 256 tasks
  int t = threadIdx.x;  // 64 entries: m1 = t>>3, m2 = t&7
  int L1, L2, L3, M3;
  long long ob, bs;
  decodeTask(r, L1, L2, L3, M3, ob, bs);
  int m1i = t >> 3, m2i = t & 7;
  float v = 0.0f;
  if (m1i <= 2 * L1 && m2i <= 2 * L2)
    v = (float)cg_coef(L1, m1i - L1, L2, m2i - L2, L3, M3 - L3);
  tab[r * 64 + t] = v;
  if (t == 0) {
    TaskMeta m;
    m.outBase = ob; m.bStride = bs;
    m.l1 = L1; m.l2 = L2; m.pad0 = 0; m.pad1 = 0;
    meta[r] = m;
  }
}

// ---------------------------------------------------------------------------
// Main kernel: one WG (4 wave32) per (b, task). Fold CG into Ahat, then
// 64x64 = Ahat^T (64xK) * x_{l2} (Kx64), K padded to 8 -> two
// v_wmma_f32_16x16x4_f32 accumulation steps per 16x16 tile.
// Store-bandwidth bound (268 MB out, > L2) -> non-temporal stores.
// ---------------------------------------------------------------------------
__global__ __launch_bounds__(128) void cg_main_kernel(
    const float* __restrict__ x0, const float* __restrict__ x1,
    const float* __restrict__ x2, const float* __restrict__ x3,
    const float* __restrict__ cgtab, const TaskMeta* __restrict__ meta,
    float* __restrict__ out) {
  __shared__ float sM[64];                      // 8x8 CG matrix [m1][m2]
  __shared__ alignas(16) float sX1[7 * 64];     // x_{l1}[b] rows
  __shared__ alignas(16) float sB[8 * 64];      // B [k][q], zero-padded K=8
  __shared__ float sA[8 * 64];                  // A [k][p] (Ahat), K=8

  const int b = blockIdx.x >> 8;   // 64 batches
  const int r = blockIdx.x & 255;  // 256 tasks

  const TaskMeta m = meta[r];      // uniform -> scalar loads
  const int n1r = 2 * m.l1 + 1;
  const int n2r = 2 * m.l2 + 1;

  const float* xs[4] = {x0, x1, x2, x3};
  const float* X1 = xs[m.l1] + (long long)b * n1r * 64;
  const float* X2 = xs[m.l2] + (long long)b * n2r * 64;

  const int tid = threadIdx.x;
  if (tid < 64) sM[tid] = cgtab[r * 64 + tid];

  // Vectorized staging: rows are 64 floats = 16 float4 each, 16B aligned.
  const float4* X1v = (const float4*)X1;
  const float4* X2v = (const float4*)X2;
  float4* sX1v = (float4*)sX1;
  float4* sBv = (float4*)sB;
  for (int i = tid; i < n1r * 16; i += 128) sX1v[i] = X1v[i];
  for (int i = tid; i < 128; i += 128) {  // 128 float4 = 512 floats
    int k = i >> 4;
    float4 z = {0.f, 0.f, 0.f, 0.f};
    sBv[i] = (k < n2r) ? X2v[i] : z;
  }
  __syncthreads();

  // Ahat[k][p] = sum_{m1} C[m1][k] * x_{l1}[b][m1][p]  (k >= n2r rows are 0
  // because padded sM columns are 0)
  for (int i = tid; i < 512; i += 128) {
    int k = i >> 6, p = i & 63;
    float s = 0.0f;
    for (int m1 = 0; m1 < n1r; ++m1) s = fmaf(sM[m1 * 8 + k], sX1[m1 * 64 + p], s);
    sA[i] = s;
  }
  __syncthreads();

  const int wave = tid >> 5;
  const int lane = tid & 31;
  const int half = lane >> 4;  // 0: lanes 0-15, 1: lanes 16-31
  const int lh = lane & 15;
  const int pt = wave;         // 16-row p-strip per wave

  // A fragments (16x4 f32): lane%16 = M row, v0/v1 = K{0,1} | K{2,3}
  const int prow = pt * 16 + lh;
  v2f a0, a1;
  a0.x = sA[(half * 2 + 0) * 64 + prow];
  a0.y = sA[(half * 2 + 1) * 64 + prow];
  a1.x = sA[(4 + half * 2 + 0) * 64 + prow];
  a1.y = sA[(4 + half * 2 + 1) * 64 + prow];

  float* outp = out + m.outBase + (long long)b * m.bStride;

#pragma unroll
  for (int qt = 0; qt < 4; ++qt) {
    const int q = qt * 16 + lh;
    v2f b0, b1;  // B (4x16 f32): lane%16 = N col, v0/v1 = K{0,1} | K{2,3}
    b0.x = sB[(half * 2 + 0) * 64 + q];
    b0.y = sB[(half * 2 + 1) * 64 + q];
    b1.x = sB[(4 + half * 2 + 0) * 64 + q];
    b1.y = sB[(4 + half * 2 + 1) * 64 + q];

    v8f c = {};
    c = __builtin_amdgcn_wmma_f32_16x16x4_f32(false, a0, false, b0, (short)0, c,
                                              false, false);
    c = __builtin_amdgcn_wmma_f32_16x16x4_f32(false, a1, false, b1, (short)0, c,
                                              false, false);

    // C/D layout: VGPR rr -> row rr (lanes<16) / row 8+rr (lanes>=16)
#pragma unroll
    for (int rr = 0; rr < 8; ++rr) {
      __builtin_nontemporal_store(c[rr],
                                  &outp[(pt * 16 + half * 8 + rr) * 64 + q]);
    }
  }
}

extern "C" void kernel_launch(void* const* d_in, const int* in_sizes, int n_in,
                              void* d_out, int out_size, void* d_ws, size_t ws_size,
                              hipStream_t stream) {
  const float* x0 = (const float*)d_in[0];
  const float* x1 = (const float*)d_in[1];
  const float* x2 = (const float*)d_in[2];
  const float* x3 = (const float*)d_in[3];

  float* tab = (float*)d_ws;                              // 64 KB CG table
  TaskMeta* meta = (TaskMeta*)((char*)d_ws + 256 * 64 * sizeof(float));  // 8 KB

  cg_init_kernel<<<256, 64, 0, stream>>>(tab, meta);
  cg_main_kernel<<<64 * 256, 128, 0, stream>>>(x0, x1, x2, x3, tab, meta,
                                               (float*)d_out);
}